// GCNGraphBaseline_19954418057675
// MI455X (gfx1250) — compile-verified
//
#include <hip/hip_runtime.h>
#include <hip/hip_bf16.h>

typedef __attribute__((ext_vector_type(2))) float v2f;
typedef __attribute__((ext_vector_type(8))) float v8f;

#define N_NODES  50000
#define N_EDGES  1600000
#define HID      128
#define N_GRAPHS 64

// ---------------- degree / normalization ----------------

__global__ __launch_bounds__(256) void k_deg_init(unsigned* __restrict__ deg, int n) {
    int i = blockIdx.x * 256 + threadIdx.x;
    if (i < n) deg[i] = 1u;  // self loop
}

__global__ __launch_bounds__(256) void k_deg_edges(const int* __restrict__ dst,
                                                   unsigned* __restrict__ deg, int ne) {
    int e = blockIdx.x * 256 + threadIdx.x;
    if (e < ne) atomicAdd(&deg[dst[e]], 1u);
}

__global__ __launch_bounds__(256) void k_dinv(float* __restrict__ dinv_f,
                                              const unsigned* __restrict__ deg, int n) {
    int i = blockIdx.x * 256 + threadIdx.x;
    if (i < n) {
        unsigned d = deg[i];
        dinv_f[i] = rsqrtf((float)d);   // deg >= 1 always (self loop)
    }
}

// ---------------- fp32 WMMA GEMM: Y = X @ W, plus self-loop init Bself = Y * dinv^2 ----
// X: [nrows,128] row-major, W: [128,128] row-major.
// 1 block = one 16-row tile; wave w (of 8) computes the 16x16 tile at columns w*16.
// NOTE: X and Bself may alias (layer 2 passes bufB for both). Block i reads only
// rows [16i,16i+16) of X and writes only those rows of Bself, and every X load
// feeds a WMMA that the epilogue stores depend on, so the ordering is safe —
// but they must NOT be declared restrict.
__global__ __launch_bounds__(256) void k_gemm128(const float* X,
                                                 const float* __restrict__ W,
                                                 const float* __restrict__ dinv,
                                                 float* __restrict__ Y,
                                                 float* Bself) {
    const int wave = threadIdx.x >> 5;
    const int lane = threadIdx.x & 31;
    const int l16  = lane & 15;
    const int h    = lane >> 4;          // half-wave select
    const int rb   = blockIdx.x * 16;    // row base
    const int cb   = wave * 16;          // col base

    v8f acc = {};
    const float* xrow = X + (size_t)(rb + l16) * HID;

    #pragma unroll
    for (int k0 = 0; k0 < HID; k0 += 4) {
        const int ka = k0 + 2 * h;                 // K for VGPR0 (VGPR1 = ka+1)
        // A 16x4: lane(l16)=M, a.x -> K=ka, a.y -> K=ka+1  (contiguous 8B load)
        v2f a = *(const v2f*)(xrow + ka);
        // B 4x16: b.x = W[ka][cb+l16], b.y = W[ka+1][cb+l16]
        v2f b;
        b.x = W[(size_t)ka * HID + cb + l16];
        b.y = W[(size_t)(ka + 1) * HID + cb + l16];
        acc = __builtin_amdgcn_wmma_f32_16x16x4_f32(
            /*neg_a=*/false, a, /*neg_b=*/false, b,
            /*c_mod=*/(short)0, acc, /*reuse_a=*/false, /*reuse_b=*/false);
    }

    // C/D layout: VGPR v, lanes 0-15 -> M=v, lanes 16-31 -> M=v+8; N = l16
    #pragma unroll
    for (int v = 0; v < 8; ++v) {
        const int row = rb + v + 8 * h;
        const float val = acc[v];
        const float di = dinv[row];
        const size_t idx = (size_t)row * HID + cb + l16;
        Y[idx] = val;                    // raw product (source of edge messages)
        Bself[idx] = val * di * di;      // self-loop contribution seeds the accumulator
    }
}

// ---------------- edge scatter: AGG[dst] += H[src] * dinv[src]*dinv[dst] ----------------
// one wave per edge; lane moves float4 (512B per wave, coalesced; resolved in L2)
__global__ __launch_bounds__(256) void k_scatter(const float* __restrict__ H,
                                                 const int* __restrict__ src,
                                                 const int* __restrict__ dst,
                                                 const float* __restrict__ dinv,
                                                 float* __restrict__ AGG, int ne) {
    const int e = blockIdx.x * 8 + (threadIdx.x >> 5);
    if (e >= ne) return;
    const int lane = threadIdx.x & 31;
    const int s = src[e];
    const int d = dst[e];
    const float nrm = dinv[s] * dinv[d];
    const float4 hv = ((const float4*)(H + (size_t)s * HID))[lane];
    float* out = AGG + (size_t)d * HID + lane * 4;
    unsafeAtomicAdd(out + 0, hv.x * nrm);
    unsafeAtomicAdd(out + 1, hv.y * nrm);
    unsafeAtomicAdd(out + 2, hv.z * nrm);
    unsafeAtomicAdd(out + 3, hv.w * nrm);
}

// ---------------- bias + relu (in place) ----------------
__global__ __launch_bounds__(256) void k_bias_relu(float* __restrict__ A,
                                                   const float* __restrict__ b, int total) {
    int i = blockIdx.x * 256 + threadIdx.x;
    if (i < total) {
        float v = A[i] + b[i & (HID - 1)];
        A[i] = v > 0.0f ? v : 0.0f;
    }
}

// ---------------- pooling ----------------

__global__ __launch_bounds__(64) void k_zero_pool(float* __restrict__ sums,
                                                  unsigned* __restrict__ cnts) {
    int i = threadIdx.x;
    sums[i] = 0.0f;
    cnts[i] = 0u;
}

// one wave per node: dot(h[i], Wh) then atomic segment accumulate
__global__ __launch_bounds__(256) void k_pool(const float* __restrict__ H,
                                              const float* __restrict__ Wh,
                                              const int* __restrict__ batch,
                                              float* __restrict__ sums,
                                              unsigned* __restrict__ cnts, int n) {
    const int i = blockIdx.x * 8 + (threadIdx.x >> 5);
    if (i >= n) return;
    const int lane = threadIdx.x & 31;
    const float4 hv = ((const float4*)(H + (size_t)i * HID))[lane];
    const float4 wv = ((const float4*)Wh)[lane];
    float d = hv.x * wv.x + hv.y * wv.y + hv.z * wv.z + hv.w * wv.w;
    #pragma unroll
    for (int off = 16; off > 0; off >>= 1)
        d += __shfl_down(d, off, 32);
    if (lane == 0) {
        const int g = batch[i];
        unsafeAtomicAdd(&sums[g], d);
        atomicAdd(&cnts[g], 1u);
    }
}

__global__ __launch_bounds__(64) void k_finalize(const float* __restrict__ sums,
                                                 const unsigned* __restrict__ cnts,
                                                 const float* __restrict__ bh,
                                                 float* __restrict__ out) {
    int g = threadIdx.x;
    float c = (float)cnts[g];
    out[g] = sums[g] / (c > 1.0f ? c : 1.0f) + bh[0];
}

// ---------------- launcher ----------------

extern "C" void kernel_launch(void* const* d_in, const int* in_sizes, int n_in,
                              void* d_out, int out_size, void* d_ws, size_t ws_size,
                              hipStream_t stream) {
    const float* x     = (const float*)d_in[0];
    const int*   c2    = (const int*)d_in[1];
    const int*   u2    = (const int*)d_in[2];
    const int*   batch = (const int*)d_in[3];
    const float* W1    = (const float*)d_in[4];
    const float* b1    = (const float*)d_in[5];
    const float* W2    = (const float*)d_in[6];
    const float* b2    = (const float*)d_in[7];
    const float* Wh    = (const float*)d_in[8];
    const float* bh    = (const float*)d_in[9];
    float* out = (float*)d_out;

    const size_t featBytes = (size_t)N_NODES * HID * sizeof(float);  // 25.6 MB
    char* ws = (char*)d_ws;
    float*    bufA = (float*)ws;                        // GEMM output (messages)
    float*    bufB = (float*)(ws + featBytes);          // aggregation accumulator
    float*    dinv = (float*)(ws + 2 * featBytes);      // also u32 degree initially
    unsigned* deg  = (unsigned*)dinv;
    float*    sums = (float*)(ws + 2 * featBytes + (size_t)N_NODES * sizeof(float));
    unsigned* cnts = (unsigned*)(sums + N_GRAPHS);

    const int nodeBlks  = (N_NODES + 255) / 256;
    const int edgeBlks  = (N_EDGES + 255) / 256;
    const int gemmBlks  = N_NODES / 16;                 // 3125 (N divisible by 16)
    const int scatBlks  = (N_EDGES + 7) / 8;            // wave per edge
    const int elemBlks  = (N_NODES * HID) / 256;
    const int poolBlks  = (N_NODES + 7) / 8;

    // normalization coefficients
    k_deg_init<<<nodeBlks, 256, 0, stream>>>(deg, N_NODES);
    k_deg_edges<<<edgeBlks, 256, 0, stream>>>(u2, deg, N_EDGES);
    k_dinv<<<nodeBlks, 256, 0, stream>>>(dinv, deg, N_NODES);

    // ---- layer 1 ----
    k_gemm128<<<gemmBlks, 256, 0, stream>>>(x, W1, dinv, bufA, bufB);
    k_scatter<<<scatBlks, 256, 0, stream>>>(bufA, c2, u2, dinv, bufB, N_EDGES);
    k_bias_relu<<<elemBlks, 256, 0, stream>>>(bufB, b1, N_NODES * HID);

    // ---- layer 2 (X and Bself alias bufB by design; see k_gemm128 note) ----
    k_gemm128<<<gemmBlks, 256, 0, stream>>>(bufB, W2, dinv, bufA, bufB);
    k_scatter<<<scatBlks, 256, 0, stream>>>(bufA, c2, u2, dinv, bufB, N_EDGES);
    k_bias_relu<<<elemBlks, 256, 0, stream>>>(bufB, b2, N_NODES * HID);

    // ---- pooling + head ----
    k_zero_pool<<<1, 64, 0, stream>>>(sums, cnts);
    k_pool<<<poolBlks, 256, 0, stream>>>(bufB, Wh, batch, sums, cnts, N_NODES);
    k_finalize<<<1, 64, 0, stream>>>(sums, cnts, bh, out);
}